// MambaClassifier_2534030705291
// MI455X (gfx1250) — compile-verified
//
#include <hip/hip_runtime.h>
#include <math.h>

// Problem constants (match reference)
#define D_MODEL 2
#define ED      64
#define NSTATE  32
#define KCONV   16
#define NC      4
#define BB      16
#define LL      2048
#define XCOLS   65          // DT_RANK + 2*N = 1 + 64
#define DBC_STRIDE 48       // cols 0..47 of w_xproj: [dt | B(32) | C0..C14(unused)]

typedef __attribute__((ext_vector_type(2))) float v2f;
typedef __attribute__((ext_vector_type(8))) float v8f;

__device__ __forceinline__ float silu_f(float x) { return x / (1.0f + __expf(-x)); }
__device__ __forceinline__ float softplus_f(float v) {
    return (v > 20.0f) ? v : log1pf(__expf(v));
}

// ---------------------------------------------------------------------------
// Kernel 1: rmsnorm -> w_in -> causal depthwise conv (K=16) -> SiLU => xb
// One 64-thread block per (b, t). Also captures z at t = L-1.
// ---------------------------------------------------------------------------
__global__ void mamba_front(const float* __restrict__ x,
                            const float* __restrict__ norm_w,
                            const float* __restrict__ w_in,     // (2, 128)
                            const float* __restrict__ conv_w,   // (64, 1, 16)
                            const float* __restrict__ conv_b,   // (64,)
                            float* __restrict__ xb,             // (B*L, 64)
                            float* __restrict__ z_last)         // (B, 64)
{
    const int blk = blockIdx.x;          // b*L + t
    const int b = blk / LL;
    const int t = blk % LL;
    const int e = threadIdx.x;           // 0..63

    __shared__ float hn[KCONV][2];       // normalized x for taps t-15..t
    if (e < KCONV) {
        const int s = t - (KCONV - 1) + e;
        float h0 = 0.0f, h1 = 0.0f;
        if (s >= 0) {
            const float x0 = x[((long)b * LL + s) * D_MODEL + 0];
            const float x1 = x[((long)b * LL + s) * D_MODEL + 1];
            const float r  = rsqrtf(0.5f * (x0 * x0 + x1 * x1) + 1e-5f);
            h0 = x0 * r * norm_w[0];
            h1 = x1 * r * norm_w[1];
        }
        hn[e][0] = h0; hn[e][1] = h1;
    }
    __syncthreads();

    const float w0 = w_in[e];            // w_in[0][e]
    const float w1 = w_in[128 + e];      // w_in[1][e]
    float acc = conv_b[e];
#pragma unroll
    for (int k = 0; k < KCONV; ++k) {
        const float pre = hn[k][0] * w0 + hn[k][1] * w1;  // == 0 for padded taps
        acc += pre * conv_w[e * KCONV + k];
    }
    xb[(long)blk * ED + e] = silu_f(acc);

    if (t == LL - 1) {
        const float z = hn[KCONV - 1][0] * w_in[ED + e]
                      + hn[KCONV - 1][1] * w_in[128 + ED + e];
        z_last[b * ED + e] = z;          // raw z; silu applied in tail
    }
}

// ---------------------------------------------------------------------------
// Kernel 2: dbc[:, 0:48] = xb @ w_xproj[:, 0:48]  via V_WMMA_F32_16X16X4_F32
// One wave (32 threads) per 16-row tile; 3 col tiles; 16 K-steps of 4.
// A layout: M = lane%16, K = vgpr + 2*(lane/16)   (ISA 7.12.2, 32-bit A 16x4)
// B layout: N = lane%16, K = vgpr + 2*(lane/16)
// C/D     : row = vgpr + 8*(lane/16), col = lane%16
// ---------------------------------------------------------------------------
__global__ void xproj_wmma(const float* __restrict__ xb,       // (B*L, 64)
                           const float* __restrict__ w_xproj,  // (64, 65) row-major
                           float* __restrict__ dbc)            // (B*L, 48)
{
    const int rt   = blockIdx.x;         // row tile over B*L rows
    const int lane = threadIdx.x;        // 0..31
    const int m    = lane & 15;
    const int hl   = lane >> 4;
    const int row0 = rt * 16;

    for (int ct = 0; ct < 3; ++ct) {
        v8f acc = {0.f, 0.f, 0.f, 0.f, 0.f, 0.f, 0.f, 0.f};
        const int col = ct * 16 + m;     // < 48 < 65, always in-range
#pragma unroll
        for (int kk = 0; kk < 16; ++kk) {
            const int kA = kk * 4 + 2 * hl;
            v2f a, bm;
            a.x  = xb[(long)(row0 + m) * ED + kA];
            a.y  = xb[(long)(row0 + m) * ED + kA + 1];
            bm.x = w_xproj[kA * XCOLS + col];
            bm.y = w_xproj[(kA + 1) * XCOLS + col];
            acc = __builtin_amdgcn_wmma_f32_16x16x4_f32(
                      /*neg_a=*/false, a, /*neg_b=*/false, bm,
                      /*c_mod=*/(short)0, acc,
                      /*reuse_a=*/false, /*reuse_b=*/false);
        }
#pragma unroll
        for (int v = 0; v < 8; ++v) {
            const int row = row0 + v + 8 * hl;
            dbc[(long)row * DBC_STRIDE + ct * 16 + m] = acc[v];
        }
    }
}

// ---------------------------------------------------------------------------
// Kernel 3: per-batch SSM scan over L, then the whole tail MLP.
// One block per batch; tid = e*16 + n0; each thread owns states (e,n0),(e,n0+16).
// ---------------------------------------------------------------------------
__global__ void mamba_scan_tail(const float* __restrict__ xb,
                                const float* __restrict__ dbc,
                                const float* __restrict__ z_last,
                                const float* __restrict__ w_xproj,
                                const float* __restrict__ w_dt,   // (1, 64)
                                const float* __restrict__ b_dt,   // (64,)
                                const float* __restrict__ A_log,  // (64, 32)
                                const float* __restrict__ D_skip, // (64,)
                                const float* __restrict__ w_out,  // (64, 64)
                                const float* __restrict__ b_out,  // (64,)
                                const float* __restrict__ w_fc,   // (64, 64)
                                const float* __restrict__ b_fc,   // (64,)
                                const float* __restrict__ w_cls,  // (64, 4)
                                const float* __restrict__ b_cls,  // (4,)
                                const float* __restrict__ w_reg,  // (64, 1)
                                const float* __restrict__ b_reg,  // (1,)
                                float* __restrict__ out)          // 64 logits + 16 mass
{
    const int b   = blockIdx.x;
    const int tid = threadIdx.x;         // 0..1023
    const int e   = tid >> 4;            // 0..63
    const int n0  = tid & 15;
    const int n1  = n0 + 16;

    const float A0  = -__expf(A_log[e * NSTATE + n0]);
    const float A1  = -__expf(A_log[e * NSTATE + n1]);
    const float wdt = w_dt[e];
    const float bdt = b_dt[e];

    float h0 = 0.0f, h1 = 0.0f;
    const long base0 = (long)b * LL;

    for (int t = 0; t < LL; ++t) {
        const long r = base0 + t;
        if (t + 8 < LL)
            __builtin_prefetch(&dbc[(r + 8) * DBC_STRIDE], 0, 1);
        const float dt0 = dbc[r * DBC_STRIDE];
        const float xv  = xb[r * ED + e];
        const float bm0 = dbc[r * DBC_STRIDE + 1 + n0];
        const float bm1 = dbc[r * DBC_STRIDE + 1 + n1];
        const float d   = softplus_f(dt0 * wdt + bdt);
        const float dx  = d * xv;
        h0 = __expf(d * A0) * h0 + dx * bm0;
        h1 = __expf(d * A1) * h1 + dx * bm1;
    }

    __shared__ float cs[NSTATE];
    __shared__ float red[1024];
    __shared__ float yy[ED];
    __shared__ float obuf[ED];
    __shared__ float hsb[ED];

    const long rl = base0 + (LL - 1);

    // C at the last timestep: C[n] = sum_e xb_last[e] * w_xproj[e, 33+n]
    if (tid < NSTATE) {
        float s = 0.0f;
        for (int ee = 0; ee < ED; ++ee)
            s += xb[rl * ED + ee] * w_xproj[ee * XCOLS + 1 + NSTATE + tid];
        cs[tid] = s;
    }
    __syncthreads();

    red[tid] = h0 * cs[n0] + h1 * cs[n1];
    __syncthreads();

    if (n0 == 0) {
        float y = 0.0f;
        for (int i = 0; i < 16; ++i) y += red[e * 16 + i];
        const float xbl = xb[rl * ED + e];
        y += D_skip[e] * xbl;
        yy[e] = y * silu_f(z_last[b * ED + e]);
    }
    __syncthreads();

    if (tid < ED) {
        float s = b_out[tid];
        for (int ee = 0; ee < ED; ++ee) s += yy[ee] * w_out[ee * ED + tid];
        obuf[tid] = s;
    }
    __syncthreads();

    if (tid < ED) {
        float s = b_fc[tid];
        for (int ee = 0; ee < ED; ++ee) s += obuf[ee] * w_fc[ee * ED + tid];
        hsb[tid] = fmaxf(s, 0.0f);
    }
    __syncthreads();

    if (tid < NC) {
        float s = b_cls[tid];
        for (int o = 0; o < ED; ++o) s += hsb[o] * w_cls[o * NC + tid];
        out[b * NC + tid] = s;                 // class_logits, row-major (B, NC)
    }
    if (tid == NC) {
        float s = b_reg[0];
        for (int o = 0; o < ED; ++o) s += hsb[o] * w_reg[o];
        out[BB * NC + b] = s;                  // mass_pred, after the 64 logits
    }
}

// ---------------------------------------------------------------------------
extern "C" void kernel_launch(void* const* d_in, const int* in_sizes, int n_in,
                              void* d_out, int out_size, void* d_ws, size_t ws_size,
                              hipStream_t stream)
{
    const float* x       = (const float*)d_in[0];
    const float* norm_w  = (const float*)d_in[1];
    const float* w_in    = (const float*)d_in[2];
    const float* conv_w  = (const float*)d_in[3];
    const float* conv_b  = (const float*)d_in[4];
    const float* w_xproj = (const float*)d_in[5];
    const float* w_dt    = (const float*)d_in[6];
    const float* b_dt    = (const float*)d_in[7];
    const float* A_log   = (const float*)d_in[8];
    const float* D_skip  = (const float*)d_in[9];
    const float* w_out   = (const float*)d_in[10];
    const float* b_out   = (const float*)d_in[11];
    const float* w_fc    = (const float*)d_in[12];
    const float* b_fc    = (const float*)d_in[13];
    const float* w_cls   = (const float*)d_in[14];
    const float* b_cls   = (const float*)d_in[15];
    const float* w_reg   = (const float*)d_in[16];
    const float* b_reg   = (const float*)d_in[17];
    float* outp = (float*)d_out;

    float* ws    = (float*)d_ws;
    float* xb    = ws;                                    // B*L*64  (8 MB)
    float* dbc   = xb  + (size_t)BB * LL * ED;            // B*L*48  (6 MB)
    float* zlast = dbc + (size_t)BB * LL * DBC_STRIDE;    // B*64

    mamba_front<<<BB * LL, ED, 0, stream>>>(x, norm_w, w_in, conv_w, conv_b,
                                            xb, zlast);
    xproj_wmma<<<(BB * LL) / 16, 32, 0, stream>>>(xb, w_xproj, dbc);
    mamba_scan_tail<<<BB, 1024, 0, stream>>>(xb, dbc, zlast, w_xproj,
                                             w_dt, b_dt, A_log, D_skip,
                                             w_out, b_out, w_fc, b_fc,
                                             w_cls, b_cls, w_reg, b_reg, outp);
}